// CombinedPriorityLoss_9740985827427
// MI455X (gfx1250) — compile-verified
//
#include <hip/hip_runtime.h>
#include <hip/hip_bf16.h>

// CDNA5 / gfx1250, wave32. WMMA f32 16x16x4 used as reduction engine.
typedef __attribute__((ext_vector_type(2))) float v2f;  // A/B: 16x4 f32 = 64 vals / 32 lanes
typedef __attribute__((ext_vector_type(8))) float v8f;  // C/D: 16x16 f32

#define TILE 256
#define MARGIN_F 0.2f

// Wave-level sum of one float per lane.
// Stage 1: V_WMMA_F32_16X16X4_F32 with A = ones(16x4), B.V0 = v, B.V1 = 0.
//   Each D column = (one low-half lane) + (one high-half lane); D is rank-1 in
//   rows, so D[0] on every lane holds colsum[lane%16].
// Stage 2: 4 x shfl_xor folds the 16 column sums -> total broadcast to all lanes.
// Requires EXEC == all 1s at the call site (block-uniform control flow only).
__device__ __forceinline__ float wave_reduce_wmma(float v) {
  v2f a; a[0] = 1.0f; a[1] = 1.0f;   // ones A-matrix
  v2f b; b[0] = v;    b[1] = 0.0f;   // partials in B row-set 0
  v8f c = {};
  c = __builtin_amdgcn_wmma_f32_16x16x4_f32(
      /*neg_a=*/false, a, /*neg_b=*/false, b,
      /*c_mod=*/(short)0, c, /*reuse_a=*/false, /*reuse_b=*/false);
  float x = c[0];                    // colsum[lane%16], both halves identical
  x += __shfl_xor(x, 1, 32);
  x += __shfl_xor(x, 2, 32);
  x += __shfl_xor(x, 4, 32);
  x += __shfl_xor(x, 8, 32);
  return x;                          // full-wave sum in every lane
}

// One block per 256x256 (i,j) tile of the pair matrix; only ti<=tj tiles work.
// Each thread owns one i and sweeps 256 j's staged in LDS.
__global__ __launch_bounds__(TILE) void rank_tiles_kernel(
    const float* __restrict__ pred, const float* __restrict__ tgt,
    float* __restrict__ partial, int N) {
  const int ti = blockIdx.y, tj = blockIdx.x;
  const int bid = ti * gridDim.x + tj;
  const int t = threadIdx.x;

  if (tj < ti) {                       // lower triangle: contribute exactly 0
    if (t == 0) partial[bid] = 0.0f;
    return;
  }

  __shared__ float sp[TILE];
  __shared__ float sg[TILE];
  __shared__ float wsum[TILE / 32];

  const int jBase = tj * TILE;
  __builtin_prefetch(pred + jBase, 0, 0);   // -> global_prefetch_b8
  __builtin_prefetch(tgt  + jBase, 0, 0);
  sp[t] = pred[jBase + t];
  sg[t] = tgt[jBase + t];

  const int i = ti * TILE + t;
  const float pi = pred[i];
  const float gi = tgt[i];
  __syncthreads();

  const bool diag = (ti == tj);
  float acc = 0.0f;
#pragma unroll 4
  for (int jj = 0; jj < TILE; ++jj) {
    const float dp  = pi - sp[jj];
    const float dt  = gi - sg[jj];
    const float hi  = fmaxf(MARGIN_F - dp, 0.0f);
    const float lo  = fmaxf(MARGIN_F + dp, 0.0f);
    const float mid = 0.1f * fabsf(dp);
    const float per = (dt > MARGIN_F) ? hi : ((dt < -MARGIN_F) ? lo : mid);
    const bool take = (!diag) || (jj > t);   // strict upper triangle on diagonal
    acc += take ? per : 0.0f;
  }

  // Deterministic block reduction: WMMA wave sum -> LDS -> serial add.
  const float w = wave_reduce_wmma(acc);
  if ((t & 31) == 0) wsum[t >> 5] = w;
  __syncthreads();
  if (t == 0) {
    float s = 0.0f;
    for (int k = 0; k < TILE / 32; ++k) s += wsum[k];
    partial[bid] = s;
  }
}

// Single block: fixed-order sums of rank partials + O(N) statistics, then the
// final scalar. Fully deterministic (no atomics anywhere).
__global__ __launch_bounds__(256) void finalize_kernel(
    const float* __restrict__ pred, const float* __restrict__ tgt,
    const float* __restrict__ partial, int nPartial,
    float* __restrict__ out, int N) {
  __shared__ float sbuf[8];
  const int t = threadIdx.x;

  float r = 0.0f;
  for (int idx = t; idx < nPartial; idx += 256) r += partial[idx];

  float ssq = 0.0f, sp = 0.0f, sp2 = 0.0f, sg = 0.0f, sg2 = 0.0f;
  for (int idx = t; idx < N; idx += 256) {
    const float p = pred[idx], g = tgt[idx];
    const float d = p - g;
    ssq += d * d;
    sp  += p;  sp2 += p * p;
    sg  += g;  sg2 += g * g;
  }

  float sums[6] = {r, ssq, sp, sp2, sg, sg2};
  float res[6];
  for (int s = 0; s < 6; ++s) {
    const float w = wave_reduce_wmma(sums[s]);
    if ((t & 31) == 0) sbuf[t >> 5] = w;
    __syncthreads();
    float tot = 0.0f;
    if (t == 0)
      for (int k = 0; k < 8; ++k) tot += sbuf[k];
    res[s] = tot;
    __syncthreads();
  }

  if (t == 0) {
    const float Nf   = (float)N;
    const float mse  = res[1] / Nf;
    const float varp = (res[3] - res[2] * res[2] / Nf) / (Nf - 1.0f);
    const float varg = (res[5] - res[4] * res[4] / Nf) / (Nf - 1.0f);
    const float dive = fmaxf(varg - varp, 0.0f);
    const float pairs = 0.5f * Nf * (Nf - 1.0f);
    const float rank = res[0] / pairs;
    out[0] = 0.1f * mse + 0.9f * rank + 0.1f * dive;
  }
}

extern "C" void kernel_launch(void* const* d_in, const int* in_sizes, int n_in,
                              void* d_out, int out_size, void* d_ws, size_t ws_size,
                              hipStream_t stream) {
  const float* pred = (const float*)d_in[0];
  const float* tgt  = (const float*)d_in[1];
  float* out        = (float*)d_out;
  const int N = in_sizes[0];                 // 8192
  const int tiles = (N + TILE - 1) / TILE;   // 32

  float* partial = (float*)d_ws;             // tiles*tiles floats (4 KB)

  dim3 grid(tiles, tiles);
  rank_tiles_kernel<<<grid, TILE, 0, stream>>>(pred, tgt, partial, N);
  finalize_kernel<<<1, 256, 0, stream>>>(pred, tgt, partial, tiles * tiles, out, N);
}